// TorchaniFeats_72378788872235
// MI455X (gfx1250) — compile-verified
//
#include <hip/hip_runtime.h>
#include <hip/hip_bf16.h>

// CDNA5 / gfx1250. fp32 WMMA path (V_WMMA_F32_16X16X4_F32), wave32.

typedef __attribute__((ext_vector_type(2))) float v2f;
typedef __attribute__((ext_vector_type(8))) float v8f;

__device__ __forceinline__ v8f wmma4(v2f a, v2f b, v8f c) {
  // 8 args: (neg_a, A, neg_b, B, c_mod, C, reuse_a, reuse_b)
  return __builtin_amdgcn_wmma_f32_16x16x4_f32(false, a, false, b, (short)0, c,
                                               false, false);
}

__device__ __forceinline__ float celu01(float x) {
  // Branchless CELU(alpha=0.1). v_exp_f32 is a single TRANS op on CDNA5;
  // 0.1*(exp(10x)-1) differs from 0.1*expm1(10x) by <~1e-8 absolute.
  float e = 0.1f * (__expf(x * 10.0f) - 1.0f);
  return x > 0.0f ? x : e;
}

struct PTable {
  const float* q[24];  // [species*6 + {W1,b1,W2,b2,W3,b3}]
};

// ---------------------------------------------------------------- utilities
__global__ void zero_counters_kernel(int* counters) {
  if (threadIdx.x < 4) counters[threadIdx.x] = 0;
}

__global__ void build_lists_kernel(const int* __restrict__ species, int nAtoms,
                                   int* __restrict__ counters,
                                   int* __restrict__ lists) {
  int i = blockIdx.x * blockDim.x + threadIdx.x;
  if (i < nAtoms) {
    int s = species[i] & 3;
    int pos = atomicAdd(&counters[s], 1);
    lists[s * nAtoms + pos] = i;
  }
}

// ------------------------------------------------------------- GEMM layers
// Y[64,N] = CELU(X[64,K] @ W^T + b), one wave handles n-tiles {wave, wave+4, ...},
// all 4 M-subtiles (M=64). X is LDS, W/b stream from global (L2-resident).
__device__ __forceinline__ void run_layer(const float* __restrict__ W,
                                          const float* __restrict__ B,
                                          const float* Xin, float* Yout, int K,
                                          int N, int wave, int l15, int hi) {
  for (int nt = wave; nt < (N >> 4); nt += 4) {
    const int n0 = nt << 4;
    v8f acc[4] = {};
    const float* wp = W + (size_t)(n0 + l15) * K + 2 * hi;
    const float* xp = Xin + (size_t)l15 * K + 2 * hi;
    for (int k0 = 0; k0 < K; k0 += 4) {
      v2f bf = *(const v2f*)(wp + k0);
#pragma unroll
      for (int mt = 0; mt < 4; ++mt) {
        v2f af = *(const v2f*)(xp + (size_t)mt * 16 * K + k0);
        acc[mt] = wmma4(af, bf, acc[mt]);
      }
    }
    const float bias = B[n0 + l15];
#pragma unroll
    for (int mt = 0; mt < 4; ++mt)
#pragma unroll
      for (int v = 0; v < 8; ++v) {
        float x = celu01(acc[mt][v] + bias);
        int row = mt * 16 + hi * 8 + v;
        Yout[(size_t)row * N + n0 + l15] = x;
      }
  }
}

// Final layer: N=96, scatter rows straight to global output.
__device__ __forceinline__ void run_layer_out(const float* __restrict__ W,
                                              const float* __restrict__ B,
                                              const float* Xin,
                                              float* __restrict__ outF,
                                              const int* idxs, int valid, int K,
                                              int wave, int l15, int hi) {
  for (int nt = wave; nt < 6; nt += 4) {
    const int n0 = nt << 4;
    v8f acc[4] = {};
    const float* wp = W + (size_t)(n0 + l15) * K + 2 * hi;
    const float* xp = Xin + (size_t)l15 * K + 2 * hi;
    for (int k0 = 0; k0 < K; k0 += 4) {
      v2f bf = *(const v2f*)(wp + k0);
#pragma unroll
      for (int mt = 0; mt < 4; ++mt) {
        v2f af = *(const v2f*)(xp + (size_t)mt * 16 * K + k0);
        acc[mt] = wmma4(af, bf, acc[mt]);
      }
    }
    const float bias = B[n0 + l15];
#pragma unroll
    for (int mt = 0; mt < 4; ++mt)
#pragma unroll
      for (int v = 0; v < 8; ++v) {
        float x = celu01(acc[mt][v] + bias);
        int row = mt * 16 + hi * 8 + v;
        if (row < valid) outF[(size_t)idxs[row] * 96 + n0 + l15] = x;
      }
  }
}

// ---------------------------------------------------------------- main MLP
__launch_bounds__(128) __global__
    void ani_mlp_kernel(const float* __restrict__ aevs,
                        const int* __restrict__ counters,
                        const int* __restrict__ lists, float* __restrict__ outF,
                        PTable pt, int nAtoms) {
  __shared__ float Xs[64 * 384];  // 96 KB
  __shared__ float Ys[64 * 160];  // 40 KB
  __shared__ int idxs[64];

  // Map blockIdx -> (species, 64-atom tile) from device-side counters.
  int t = blockIdx.x;
  int s, cnt = 0;
  for (s = 0; s < 4; ++s) {
    cnt = counters[s];
    int tiles = (cnt + 63) >> 6;
    if (t < tiles) break;
    t -= tiles;
  }
  if (s >= 4) return;

  const int D1v[4] = {160, 144, 128, 128};
  const int D2v[4] = {128, 112, 112, 112};
  const int d1 = D1v[s], d2 = D2v[s];
  const float* W1 = pt.q[s * 6 + 0];
  const float* b1 = pt.q[s * 6 + 1];
  const float* W2 = pt.q[s * 6 + 2];
  const float* b2 = pt.q[s * 6 + 3];
  const float* W3 = pt.q[s * 6 + 4];
  const float* b3 = pt.q[s * 6 + 5];

  const int tid = threadIdx.x;
  const int base = t * 64;
  int valid = cnt - base;
  if (valid > 64) valid = 64;

  if (tid < 64) {
    int m = base + tid;
    if (m >= cnt) m = cnt - 1;  // pad with a safe (valid) index
    idxs[tid] = lists[s * nAtoms + m];
  }
  __syncthreads();

  // Gather X: 64 rows x 384 f32, float4 granularity (96 float4 per row).
  for (int i = tid; i < 64 * 96; i += 128) {
    int row = i / 96, c4 = i - row * 96;
    int atom = idxs[row];
    ((float4*)Xs)[row * 96 + c4] =
        ((const float4*)(aevs + (size_t)atom * 384))[c4];
  }
  __syncthreads();

  const int lane = tid & 31;
  const int wave = tid >> 5;
  const int l15 = lane & 15;
  const int hi = lane >> 4;

  run_layer(W1, b1, Xs, Ys, 384, d1, wave, l15, hi);
  __syncthreads();
  run_layer(W2, b2, Ys, Xs, d1, d2, wave, l15, hi);
  __syncthreads();
  run_layer_out(W3, b3, Xs, outF, idxs, valid, d2, wave, l15, hi);
}

// ------------------------------------------------------------------- launch
extern "C" void kernel_launch(void* const* d_in, const int* in_sizes, int n_in,
                              void* d_out, int out_size, void* d_ws,
                              size_t ws_size, hipStream_t stream) {
  const int* species = (const int*)d_in[0];
  const float* aevs = (const float*)d_in[1];
  const int nAtoms = in_sizes[0];  // 512*128 = 65536

  // ws layout: [0..15] counters (4 ints), then 4 * nAtoms index lists.
  int* counters = (int*)d_ws;
  int* lists = (int*)((char*)d_ws + 16);

  // Output tuple: (species bits, features). Copy species verbatim.
  hipMemcpyAsync(d_out, d_in[0], (size_t)nAtoms * sizeof(int),
                 hipMemcpyDeviceToDevice, stream);

  zero_counters_kernel<<<1, 32, 0, stream>>>(counters);
  build_lists_kernel<<<(nAtoms + 255) / 256, 256, 0, stream>>>(
      species, nAtoms, counters, lists);

  PTable pt;
  for (int i = 0; i < 24; ++i) pt.q[i] = (const float*)d_in[2 + i];

  float* outF = (float*)d_out + nAtoms;
  const int maxTiles = nAtoms / 64 + 4;  // upper bound on sum of ceil(c_s/64)
  ani_mlp_kernel<<<maxTiles, 128, 0, stream>>>(aevs, counters, lists, outF, pt,
                                               nAtoms);
}